// gatedPool_l_24575802867832
// MI455X (gfx1250) — compile-verified
//
#include <hip/hip_runtime.h>

// Gated pool: out = sigmoid(conv2x2(x, mask)) * maxpool2x2(x)
//           + (1 - sigmoid(...)) * avgpool2x2(x),  stride 2, VALID.
// x: (32,128,112,112) f32, mask: (1,1,2,2) f32, out: (32,128,56,56) f32.
//
// HBM-bound: 257 MB / 23.3 TB/s ~= 11 us. One streaming pass; 4-tap conv +
// avg folded into V_WMMA_F32_16X16X4_F32 (B = window data, A = [mask; 0.25]).
// Grid-stride loop (~16 groups/wave) amortizes prologue; fast-rcp sigmoid
// keeps per-group VALU low so instruction issue stays under the HBM time.

typedef __attribute__((ext_vector_type(2))) float v2f;
typedef __attribute__((ext_vector_type(8))) float v8f;

#define BC_   4096   // B*C = 32*128
#define H_    112
#define W_    112
#define HO_   56
#define WO_   56
#define WPB_  8      // wave32 per block
#define NBLK_ 6144   // 49,152 waves; ~16 groups each

__global__ __launch_bounds__(256) void gated_pool_wmma(
    const float* __restrict__ x, const float* __restrict__ mask,
    float* __restrict__ out, int ngroups)
{
  const int lane = threadIdx.x & 31;
  const int wave = threadIdx.x >> 5;
  const int lm   = lane & 15;                  // pixel slot within group
  const int half = lane >> 4;                  // 0: window row 0, 1: window row 1

  // A matrix 16x4 f32 (built once, lives in 2 VGPRs for the whole kernel):
  // row0 = [m0 m1 m2 m3], row1 = [.25 .25 .25 .25], rest 0.
  // f32 A layout: lane m holds A[m,0..1]; lane m+16 holds A[m,2..3].
  const float m0 = mask[0], m1 = mask[1], m2 = mask[2], m3 = mask[3];
  v2f a;
  a.x = (lm == 0) ? (half ? m2 : m0) : ((lm == 1) ? 0.25f : 0.0f);
  a.y = (lm == 0) ? (half ? m3 : m1) : ((lm == 1) ? 0.25f : 0.0f);

  const int gstride = gridDim.x * WPB_;
  for (int g = blockIdx.x * WPB_ + wave; g < ngroups; g += gstride) {
    // Output pixel handled by this lane-pair (lanes lm and lm+16 share p).
    const int p  = g * 16 + lm;
    const int wo = p % WO_;
    const int t  = p / WO_;
    const int ho = t % HO_;
    const int bc = t / HO_;

    // Lanes 0-15: row-(2*ho) float2 of 16 consecutive pixels (128B);
    // lanes 16-31: row-(2*ho+1) float2 (128B). Fully coalesced, NT (no reuse,
    // 205MB stream > 192MB L2).
    const long in_off = (long)bc * (H_ * W_) + (long)(2 * ho + half) * W_ + 2 * wo;
    const v2f pair = __builtin_nontemporal_load((const v2f*)(x + in_off));

    // B matrix 4x16 f32: column n = window of pixel n = [x0 x1 y0 y1].
    // Lane n holds B[0..1,n]; lane n+16 holds B[2..3,n] -- i.e. exactly `pair`.
    v8f c = {};
    // D[0,n] = gate(n), D[1,n] = avg(n); land in d[0], d[1] of lanes 0-15.
    v8f d = __builtin_amdgcn_wmma_f32_16x16x4_f32(
        /*neg_a=*/false, a, /*neg_b=*/false, pair,
        /*c_mod=*/(short)0, c, /*reuse_a=*/false, /*reuse_b=*/false);

    // Max-pool: combine this half-window's max with the other half (lane^16).
    float mhalf = fmaxf(pair.x, pair.y);
    float mx    = fmaxf(mhalf, __shfl_xor(mhalf, 16));

    if (half == 0) {
      float gate  = d[0];
      float av    = d[1];
      // sigmoid via single v_exp_f32 + v_rcp_f32 (no IEEE divide sequence)
      float e     = __builtin_amdgcn_exp2f(gate * -1.44269504088896340736f);
      float alpha = __builtin_amdgcn_rcpf(1.0f + e);
      float res   = fmaf(alpha, mx - av, av);  // alpha*mx + (1-alpha)*av
      __builtin_nontemporal_store(res, out + p);  // 64B coalesced NT store
    }
  }
}

extern "C" void kernel_launch(void* const* d_in, const int* in_sizes, int n_in,
                              void* d_out, int out_size, void* d_ws, size_t ws_size,
                              hipStream_t stream) {
  const float* x    = (const float*)d_in[0];
  const float* mask = (const float*)d_in[1];
  float* out        = (float*)d_out;

  const int npix    = BC_ * HO_ * WO_;   // 12,845,056
  const int ngroups = npix / 16;         // 802,816 (exact)
  int nblk = NBLK_;
  int maxblk = (ngroups + WPB_ - 1) / WPB_;
  if (nblk > maxblk) nblk = maxblk;
  gated_pool_wmma<<<dim3(nblk), dim3(256), 0, stream>>>(x, mask, out, ngroups);
}